// GAT_61692910240233
// MI455X (gfx1250) — compile-verified
//
#include <hip/hip_runtime.h>
#include <hip/hip_bf16.h>
#include <math.h>

// ---------------------------------------------------------------------------
// GATv2 GNN forward for MI455X (gfx1250, wave32).
// Dense node GEMMs: V_WMMA_F32_16X16X4_F32 with weights staged in LDS.
// Edge softmax/scatter: f32 global atomics with We/att staged in LDS.
// ---------------------------------------------------------------------------

typedef __attribute__((ext_vector_type(2))) float v2f;
typedef __attribute__((ext_vector_type(8))) float v8f;

#define TPB 256
#define WAVES_PER_BLOCK 8

static constexpr int F_IN    = 128;
static constexpr int ED      = 16;
static constexpr int HID     = 32;
static constexpr int HEADS   = 2;
static constexpr int D1      = HID * HEADS;  // 64
static constexpr int NGRAPH  = 64;
static constexpr int NCLS    = 2;
static constexpr float BN_EPS = 1e-5f;

__device__ __forceinline__ void atomicMaxF(float* addr, float val) {
  // monotone float<->int trick; works with -inf initialization
  if (val >= 0.0f) atomicMax(reinterpret_cast<int*>(addr), __float_as_int(val));
  else             atomicMin(reinterpret_cast<unsigned int*>(addr), __float_as_uint(val));
}

// ---------------------------------------------------------------------------
// C[rows,K] = A[rows,M] @ W[M,K] (+ bias).  rows%16==0, K%16==0, M%4==0.
// Entire W (<=32KB) is staged into LDS once per block; one wave per 16x16
// output tile; reduction in steps of 4 via f32 WMMA.
//   A frag (16x4):  lane l holds A[tr*16 + l%16][m0 + (l/16)*2 + {0,1}]
//   B frag (4x16):  lane l holds W[m0 + (l/16)*2 + {0,1}][tc*16 + l%16]
//   C/D (16x16):    VGPR v, lane l -> C[tr*16 + (l/16)*8 + v][tc*16 + l%16]
// ---------------------------------------------------------------------------
__global__ void gemm_wmma_f32(const float* __restrict__ A,
                              const float* __restrict__ W,
                              const float* __restrict__ bias,
                              float* __restrict__ C,
                              int rows, int M, int K) {
  extern __shared__ float sW[];                 // M*K floats
  const int total = M * K;
  for (int i = threadIdx.x * 4; i < total; i += blockDim.x * 4) {
    *(float4*)&sW[i] = *(const float4*)&W[i];   // coalesced 16B staging
  }
  __syncthreads();

  const int lane = threadIdx.x & 31;
  const int wave = threadIdx.x >> 5;
  const int tilesPerCol = K >> 4;
  const int nTiles = (rows >> 4) * tilesPerCol;
  const int tile = blockIdx.x * WAVES_PER_BLOCK + wave;
  if (tile >= nTiles) return;  // wave-uniform: EXEC stays all-ones for WMMA

  const int tr = tile / tilesPerCol;
  const int tc = tile - tr * tilesPerCol;
  const int lmod = lane & 15;
  const int lhi  = lane >> 4;          // 0 or 1
  const int kk   = lhi * 2;            // K-sub-offset within each 4-step
  const int col  = tc * 16 + lmod;     // B/C column for this lane

  const float* __restrict__ Arow = A + (size_t)(tr * 16 + lmod) * M;
  v8f acc = {};
  for (int m0 = 0; m0 < M; m0 += 4) {
    v2f a = *(const v2f*)(Arow + m0 + kk);      // 8B aligned: kk even, row 8B-aligned
    v2f b;
    b.x = sW[(m0 + kk + 0) * K + col];
    b.y = sW[(m0 + kk + 1) * K + col];
    acc = __builtin_amdgcn_wmma_f32_16x16x4_f32(false, a, false, b,
                                                (short)0, acc, false, false);
  }
  const float bv = bias ? bias[col] : 0.0f;
  const int rbase = tr * 16 + lhi * 8;
#pragma unroll
  for (int v = 0; v < 8; ++v) {
    C[(size_t)(rbase + v) * K + col] = acc[v] + bv;
  }
}

// ---------------------------------------------------------------------------
__global__ void fill_f32(float* __restrict__ p, int n, float v) {
  int i = blockIdx.x * blockDim.x + threadIdx.x;
  int stride = gridDim.x * blockDim.x;
  for (; i < n; i += stride) p[i] = v;
}

// score[e,h] = sum_c leaky(xl[src,h,c] + xr[dst,h,c] + (ea[e]@We)[h,c]) * att[h,c]
// also: smax[dst,h] = max(...).  We/att staged in LDS (shared by all threads).
__global__ void gat_score(const float* __restrict__ xl, const float* __restrict__ xr,
                          const float* __restrict__ ea, const float* __restrict__ We,
                          const float* __restrict__ att,
                          const int* __restrict__ src, const int* __restrict__ dst,
                          float* __restrict__ score, float* __restrict__ smax,
                          int E, int H, int C) {
  __shared__ float sWe[ED * 64];
  __shared__ float sAtt[64];
  const int HC = H * C;
  for (int i = threadIdx.x; i < ED * HC; i += blockDim.x) sWe[i] = We[i];
  if ((int)threadIdx.x < HC) sAtt[threadIdx.x] = att[threadIdx.x];
  __syncthreads();

  const int t = blockIdx.x * blockDim.x + threadIdx.x;
  if (t >= E * H) return;
  const int e = t / H, h = t - e * H;
  const int s = src[e], d = dst[e];
  const float* __restrict__ xls = xl + (size_t)s * HC + h * C;
  const float* __restrict__ xrd = xr + (size_t)d * HC + h * C;
  const float* __restrict__ ear = ea + (size_t)e * ED;
  const float* sWec = sWe + h * C;              // stride HC between rows
  const float* sath = sAtt + h * C;

  float eb[ED];
#pragma unroll
  for (int j = 0; j < ED; j += 4) {
    float4 v = *(const float4*)&ear[j];
    eb[j] = v.x; eb[j + 1] = v.y; eb[j + 2] = v.z; eb[j + 3] = v.w;
  }

  float sc = 0.0f;
  for (int c4 = 0; c4 < C; c4 += 4) {
    float4 xa = *(const float4*)&xls[c4];       // 16B aligned: h*C mult of 32
    float4 xb = *(const float4*)&xrd[c4];
    float am[4] = {xa.x + xb.x, xa.y + xb.y, xa.z + xb.z, xa.w + xb.w};
#pragma unroll
    for (int u = 0; u < 4; ++u) {
      const int c = c4 + u;
      float ef = 0.0f;
#pragma unroll
      for (int j = 0; j < ED; ++j) ef = fmaf(eb[j], sWec[j * HC + c], ef);
      float m = am[u] + ef;
      m = (m > 0.0f) ? m : 0.2f * m;            // LeakyReLU(0.2)
      sc = fmaf(m, sath[c], sc);
    }
  }
  score[t] = sc;
  atomicMaxF(&smax[d * H + h], sc);
}

// ex = exp(score - smax[dst]); denom[dst] += ex
__global__ void gat_expsum(const float* __restrict__ smax,
                           const int* __restrict__ dst,
                           float* __restrict__ score, float* __restrict__ denom,
                           int E, int H) {
  const int t = blockIdx.x * blockDim.x + threadIdx.x;
  if (t >= E * H) return;
  const int e = t / H, h = t - e * H;
  const int d = dst[e];
  const float ex = expf(score[t] - smax[d * H + h]);
  score[t] = ex;
  atomicAdd(&denom[d * H + h], ex);
}

// out[dst,h,c] += xl[src,h,c] * (ex / (denom[dst,h] + 1e-16))
__global__ void gat_scatter(const float* __restrict__ xl,
                            const float* __restrict__ score,
                            const float* __restrict__ denom,
                            const int* __restrict__ src, const int* __restrict__ dst,
                            float* __restrict__ out, int E, int H, int C) {
  const int t = blockIdx.x * blockDim.x + threadIdx.x;
  if (t >= E * H) return;
  const int e = t / H, h = t - e * H;
  const int s = src[e], d = dst[e];
  const int HC = H * C;
  const float alpha = score[t] / (denom[d * H + h] + 1e-16f);
  const float* __restrict__ xls = xl + (size_t)s * HC + h * C;
  float* __restrict__ o = out + (size_t)d * HC + h * C;
  for (int c4 = 0; c4 < C; c4 += 4) {
    float4 v = *(const float4*)&xls[c4];
    atomicAdd(&o[c4 + 0], v.x * alpha);
    atomicAdd(&o[c4 + 1], v.y * alpha);
    atomicAdd(&o[c4 + 2], v.z * alpha);
    atomicAdd(&o[c4 + 3], v.w * alpha);
  }
}

// h = elu(batchnorm(acc + conv_bias) + xp)
__global__ void bn_elu_skip(const float* __restrict__ acc,
                            const float* __restrict__ cbias,
                            const float* __restrict__ gamma, const float* __restrict__ beta,
                            const float* __restrict__ mean, const float* __restrict__ var,
                            const float* __restrict__ xp,
                            float* __restrict__ out, int n, int D) {
  const int i = blockIdx.x * blockDim.x + threadIdx.x;
  if (i >= n) return;
  const int f = i % D;
  float v = acc[i] + cbias[f];
  v = (v - mean[f]) * gamma[f] * rsqrtf(var[f] + BN_EPS) + beta[f];
  v += xp[i];
  out[i] = (v > 0.0f) ? v : expm1f(v);
}

// per-graph sum / max / count
__global__ void pool_kernel(const float* __restrict__ h2, const int* __restrict__ batch,
                            float* __restrict__ gsum, float* __restrict__ gmax,
                            float* __restrict__ gcnt, int n, int D) {
  const int t = blockIdx.x * blockDim.x + threadIdx.x;
  if (t >= n * D) return;
  const int node = t / D, f = t - node * D;
  const int g = batch[node];
  const float v = h2[t];
  atomicAdd(&gsum[g * D + f], v);
  atomicMaxF(&gmax[g * D + f], v);
  if (f == 0) atomicAdd(&gcnt[g], 1.0f);
}

// out[g,c] = [mean|max] @ cls_W + cls_b
__global__ void classify_kernel(const float* __restrict__ gsum, const float* __restrict__ gmax,
                                const float* __restrict__ gcnt,
                                const float* __restrict__ Wc, const float* __restrict__ bc,
                                float* __restrict__ out, int ng, int D, int ncls) {
  const int t = blockIdx.x * blockDim.x + threadIdx.x;
  if (t >= ng * ncls) return;
  const int g = t / ncls, c = t - g * ncls;
  const float cnt = fmaxf(gcnt[g], 1.0f);
  float acc = bc[c];
  for (int f = 0; f < D; ++f)
    acc = fmaf(gsum[g * D + f] / cnt, Wc[(size_t)f * ncls + c], acc);
  for (int f = 0; f < D; ++f) {
    float mx = gmax[g * D + f];
    if (!(mx > -3.0e38f)) mx = 0.0f;  // -inf / NaN -> 0 (matches reference)
    acc = fmaf(mx, Wc[(size_t)(D + f) * ncls + c], acc);
  }
  out[t] = acc;
}

// ---------------------------------------------------------------------------
static inline int cdiv(long long a, long long b) { return (int)((a + b - 1) / b); }

extern "C" void kernel_launch(void* const* d_in, const int* in_sizes, int n_in,
                              void* d_out, int out_size, void* d_ws, size_t ws_size,
                              hipStream_t stream) {
  // ---- inputs (setup_inputs dict order) ----
  const float* x         = (const float*)d_in[0];
  const float* edge_attr = (const float*)d_in[1];
  const int*   eidx      = (const int*)d_in[2];
  const int*   batch     = (const int*)d_in[3];
  const float* c1_Wl  = (const float*)d_in[4];
  const float* c1_bl  = (const float*)d_in[5];
  const float* c1_Wr  = (const float*)d_in[6];
  const float* c1_br  = (const float*)d_in[7];
  const float* c1_We  = (const float*)d_in[8];
  const float* c1_att = (const float*)d_in[9];
  const float* c1_bias= (const float*)d_in[10];
  const float* bn1_g  = (const float*)d_in[11];
  const float* bn1_b  = (const float*)d_in[12];
  const float* bn1_m  = (const float*)d_in[13];
  const float* bn1_v  = (const float*)d_in[14];
  const float* sp1_W  = (const float*)d_in[15];
  const float* sp1_b  = (const float*)d_in[16];
  const float* c2_Wl  = (const float*)d_in[17];
  const float* c2_bl  = (const float*)d_in[18];
  const float* c2_Wr  = (const float*)d_in[19];
  const float* c2_br  = (const float*)d_in[20];
  const float* c2_We  = (const float*)d_in[21];
  const float* c2_att = (const float*)d_in[22];
  const float* c2_bias= (const float*)d_in[23];
  const float* bn2_g  = (const float*)d_in[24];
  const float* bn2_b  = (const float*)d_in[25];
  const float* bn2_m  = (const float*)d_in[26];
  const float* bn2_v  = (const float*)d_in[27];
  const float* sp2_W  = (const float*)d_in[28];
  const float* sp2_b  = (const float*)d_in[29];
  const float* cls_W  = (const float*)d_in[30];
  const float* cls_b  = (const float*)d_in[31];

  const int N = in_sizes[0] / F_IN;       // 50000
  const int E = in_sizes[2] / 2;          // 1600000
  const int* src = eidx;                  // edge_index[0]
  const int* dst = eidx + E;              // edge_index[1]

  // ---- workspace carving ----
  float* p = (float*)d_ws;
  auto take = [&](size_t nElem) {
    float* r = p;
    p += (nElem + 63) & ~(size_t)63;      // 256B alignment
    return r;
  };
  float* xl1   = take((size_t)N * D1);
  float* xr1   = take((size_t)N * D1);
  float* xp1   = take((size_t)N * D1);
  float* out1  = take((size_t)N * D1);
  float* h1    = take((size_t)N * D1);
  float* sc    = take((size_t)E * HEADS);   // reused (E*1) for conv2
  float* smax  = take((size_t)N * HEADS);
  float* denom = take((size_t)N * HEADS);
  float* xl2   = take((size_t)N * HID);
  float* xr2   = take((size_t)N * HID);
  float* xp2   = take((size_t)N * HID);
  float* out2  = take((size_t)N * HID);
  float* h2    = take((size_t)N * HID);
  float* gsum  = take((size_t)NGRAPH * HID);
  float* gmax  = take((size_t)NGRAPH * HID);
  float* gcnt  = take((size_t)NGRAPH);
  (void)ws_size;

  const float NEG_INF = -__builtin_inff();

  // ---- layer 1: node GEMMs (WMMA, W staged in LDS) ----
  {
    int tiles = (N / 16) * (D1 / 16);
    int blks = cdiv(tiles, WAVES_PER_BLOCK);
    size_t shmem = (size_t)F_IN * D1 * sizeof(float);   // 32KB
    gemm_wmma_f32<<<blks, TPB, shmem, stream>>>(x, c1_Wl, c1_bl, xl1, N, F_IN, D1);
    gemm_wmma_f32<<<blks, TPB, shmem, stream>>>(x, c1_Wr, c1_br, xr1, N, F_IN, D1);
    gemm_wmma_f32<<<blks, TPB, shmem, stream>>>(x, sp1_W, sp1_b, xp1, N, F_IN, D1);
  }
  // ---- layer 1: edge softmax-attention ----
  fill_f32<<<cdiv((size_t)N * HEADS, TPB), TPB, 0, stream>>>(smax, N * HEADS, NEG_INF);
  fill_f32<<<cdiv((size_t)N * HEADS, TPB), TPB, 0, stream>>>(denom, N * HEADS, 0.0f);
  fill_f32<<<cdiv((size_t)N * D1, TPB), TPB, 0, stream>>>(out1, N * D1, 0.0f);
  {
    int nt = E * HEADS;
    gat_score<<<cdiv(nt, TPB), TPB, 0, stream>>>(xl1, xr1, edge_attr, c1_We, c1_att,
                                                 src, dst, sc, smax, E, HEADS, HID);
    gat_expsum<<<cdiv(nt, TPB), TPB, 0, stream>>>(smax, dst, sc, denom, E, HEADS);
    gat_scatter<<<cdiv(nt, TPB), TPB, 0, stream>>>(xl1, sc, denom, src, dst, out1,
                                                   E, HEADS, HID);
  }
  bn_elu_skip<<<cdiv((size_t)N * D1, TPB), TPB, 0, stream>>>(
      out1, c1_bias, bn1_g, bn1_b, bn1_m, bn1_v, xp1, h1, N * D1, D1);

  // ---- layer 2: node GEMMs (WMMA, W staged in LDS) ----
  {
    int tiles = (N / 16) * (HID / 16);
    int blks = cdiv(tiles, WAVES_PER_BLOCK);
    size_t shmem = (size_t)D1 * HID * sizeof(float);    // 8KB
    gemm_wmma_f32<<<blks, TPB, shmem, stream>>>(h1, c2_Wl, c2_bl, xl2, N, D1, HID);
    gemm_wmma_f32<<<blks, TPB, shmem, stream>>>(h1, c2_Wr, c2_br, xr2, N, D1, HID);
    gemm_wmma_f32<<<blks, TPB, shmem, stream>>>(h1, sp2_W, sp2_b, xp2, N, D1, HID);
  }
  // ---- layer 2: edge softmax-attention (H=1) ----
  fill_f32<<<cdiv(N, TPB), TPB, 0, stream>>>(smax, N, NEG_INF);
  fill_f32<<<cdiv(N, TPB), TPB, 0, stream>>>(denom, N, 0.0f);
  fill_f32<<<cdiv((size_t)N * HID, TPB), TPB, 0, stream>>>(out2, N * HID, 0.0f);
  {
    gat_score<<<cdiv(E, TPB), TPB, 0, stream>>>(xl2, xr2, edge_attr, c2_We, c2_att,
                                                src, dst, sc, smax, E, 1, HID);
    gat_expsum<<<cdiv(E, TPB), TPB, 0, stream>>>(smax, dst, sc, denom, E, 1);
    gat_scatter<<<cdiv(E, TPB), TPB, 0, stream>>>(xl2, sc, denom, src, dst, out2,
                                                  E, 1, HID);
  }
  bn_elu_skip<<<cdiv((size_t)N * HID, TPB), TPB, 0, stream>>>(
      out2, c2_bias, bn2_g, bn2_b, bn2_m, bn2_v, xp2, h2, N * HID, HID);

  // ---- pooling + classifier ----
  fill_f32<<<cdiv(NGRAPH * HID, TPB), TPB, 0, stream>>>(gsum, NGRAPH * HID, 0.0f);
  fill_f32<<<cdiv(NGRAPH * HID, TPB), TPB, 0, stream>>>(gmax, NGRAPH * HID, NEG_INF);
  fill_f32<<<1, TPB, 0, stream>>>(gcnt, NGRAPH, 0.0f);
  pool_kernel<<<cdiv((size_t)N * HID, TPB), TPB, 0, stream>>>(h2, batch, gsum, gmax,
                                                              gcnt, N, HID);
  classify_kernel<<<1, TPB, 0, stream>>>(gsum, gmax, gcnt, cls_W, cls_b,
                                         (float*)d_out, NGRAPH, NCLS == 2 ? HID : HID, NCLS);
  (void)out_size; (void)n_in;
}